// EnergyDistributionCNN_43576738185708
// MI455X (gfx1250) — compile-verified
//
#include <hip/hip_runtime.h>
#include <cmath>

// ---------------------------------------------------------------------------
// EnergyDistributionCNN, algebraically fused:
//   out = conv3x3(x,k) ; E = exp(out) (E=1 at OOB, i.e. exp(0) of the zero pad)
//   Z   = 3x3 boxsum(E) ; Wq = x / Z (0 at OOB)
//   y   = E * 3x3 boxsum(Wq)
// Single pass over HBM (134 MB min traffic -> ~5.8us roofline @ 23.3 TB/s).
// Interior tiles are staged into LDS by two parallel TDM tile DMAs
// (one per wave; TDM ops from different waves are unordered -> overlap).
// ---------------------------------------------------------------------------

typedef unsigned int v4u __attribute__((ext_vector_type(4)));
typedef int          v4i __attribute__((ext_vector_type(4)));
typedef int          v8i __attribute__((ext_vector_type(8)));
typedef float        f4  __attribute__((ext_vector_type(4)));

#define TILE   64
#define XS_W   70   // TILE + 2*3 halo
#define ES_W   68   // TILE + 2*2
#define WS_W   66   // TILE + 2*1
#define HALF_ROWS 35  // 70 rows split across two TDM descriptors

__global__ __launch_bounds__(256)
void energy_dist_fused_kernel(const float* __restrict__ x,
                              const float* __restrict__ kern,
                              float* __restrict__ y,
                              int H, int W) {
    const int tid  = threadIdx.x;
    const int row0 = blockIdx.y * TILE;
    const int col0 = blockIdx.x * TILE;

    __shared__ float xs[XS_W * XS_W];   // x tile, halo 3  (19600 B)
    __shared__ float es[ES_W * ES_W];   // E = exp(conv)   (18496 B)
    __shared__ float ws[WS_W * WS_W];   // Wq = x / Z      (17424 B)

    // 3x3 kernel into registers (uniform)
    float kv[9];
#pragma unroll
    for (int i = 0; i < 9; ++i) kv[i] = kern[i];

    // ---- Stage 1: load 70x70 x-tile (global -> LDS) -----------------------
    const bool interior = (row0 >= 3) && (col0 >= 3) &&
                          (row0 + XS_W - 3 <= H) && (col0 + XS_W - 3 <= W);

    if (interior) {
        // Fully in-bounds: two TDM 2D tile DMAs (35 rows each), one issued by
        // wave 0 and one by wave 1 -> the DMAs run in parallel.
        const int wave = tid >> 5;
        if (wave < 2) {
            const int rbeg = wave * HALF_ROWS;
            unsigned long long ga = (unsigned long long)(const void*)
                (x + (size_t)(row0 - 3 + rbeg) * W + (col0 - 3));
            // Generic LDS pointer: low 32 bits are the DS-space byte offset.
            unsigned lds_off = (unsigned)(unsigned long long)(const void*)xs
                             + (unsigned)(rbeg * XS_W * sizeof(float));

            // D# group 0: count=1 | lds_addr | global_addr[56:0] | type=2
            v4u g0 = { 1u,
                       lds_off,
                       (unsigned)ga,
                       ((unsigned)(ga >> 32) & 0x01FFFFFFu) | 0x80000000u };
            // D# group 1: data_size=4B; tensor view 70x35; tile 70x35; stride=W
            v8i g1 = { (int)(2u << 16),                 // data_size = 2 (4 B)
                       (int)((unsigned)XS_W << 16),     // tensor_dim0[15:0]
                       (int)((unsigned)HALF_ROWS << 16),// tensor_dim1[15:0]
                       (int)((unsigned)XS_W << 16),     // tile_dim0
                       HALF_ROWS,                       // tile_dim1
                       W,                               // tensor_dim0_stride
                       0, 0 };
            v4i gz = { 0, 0, 0, 0 };
#if __has_include(<hip/amd_detail/amd_gfx1250_TDM.h>)
            v8i z8 = { 0, 0, 0, 0, 0, 0, 0, 0 };
            __builtin_amdgcn_tensor_load_to_lds(g0, g1, gz, gz, z8, 0);
#else
            __builtin_amdgcn_tensor_load_to_lds(g0, g1, gz, gz, 0);
#endif
            __builtin_amdgcn_s_wait_tensorcnt(0);
        }
    } else {
        // Border tile: guarded loads; OOB -> 0 (== conv zero padding).
        for (int i = tid; i < XS_W * XS_W; i += 256) {
            int r  = i / XS_W, c = i % XS_W;
            int gr = row0 - 3 + r, gc = col0 - 3 + c;
            float v = 0.0f;
            if ((unsigned)gr < (unsigned)H && (unsigned)gc < (unsigned)W)
                v = x[(size_t)gr * W + gc];
            xs[i] = v;
        }
    }
    __syncthreads();

    // ---- Stage 2: E = exp(conv3x3(x)) on 68x68 (E=1 where globally OOB) ---
    for (int i = tid; i < ES_W * ES_W; i += 256) {
        int r  = i / ES_W, c = i % ES_W;
        int gr = row0 - 2 + r, gc = col0 - 2 + c;
        float e = 1.0f;   // exp(0): zero-pad logits participate in softmax
        if ((unsigned)gr < (unsigned)H && (unsigned)gc < (unsigned)W) {
            const float* p = &xs[r * XS_W + c];
            float acc = p[0]        * kv[0] + p[1]          * kv[1] + p[2]          * kv[2]
                      + p[XS_W]     * kv[3] + p[XS_W + 1]   * kv[4] + p[XS_W + 2]   * kv[5]
                      + p[2 * XS_W] * kv[6] + p[2*XS_W + 1] * kv[7] + p[2*XS_W + 2] * kv[8];
            e = __expf(acc);
        }
        es[i] = e;
    }
    __syncthreads();

    // ---- Stage 3: Wq = x / Z, Z = 3x3 boxsum(E), on 66x66 -----------------
    for (int i = tid; i < WS_W * WS_W; i += 256) {
        int r = i / WS_W, c = i % WS_W;
        const float* p = &es[r * ES_W + c];
        float z = p[0]        + p[1]          + p[2]
                + p[ES_W]     + p[ES_W + 1]   + p[ES_W + 2]
                + p[2 * ES_W] + p[2*ES_W + 1] + p[2*ES_W + 2];
        float xv = xs[(r + 2) * XS_W + (c + 2)];   // 0 where globally OOB
        ws[i] = xv * __builtin_amdgcn_rcpf(z);     // Z > 0 always (sum of exps)
    }
    __syncthreads();

    // ---- Stage 4: y = E * 3x3 boxsum(Wq); 4 outputs/thread, b128 stores ---
    // 64x64 outputs as 1024 quads: r = i>>4, c0 = (i&15)*4 (shift/mask only).
    for (int i = tid; i < (TILE * TILE) / 4; i += 256) {
        int r  = i >> 4;
        int c0 = (i & 15) << 2;
        const float* p = &ws[r * WS_W + c0];
        // sliding-window column sums over 6 columns x 3 rows
        float cs[6];
#pragma unroll
        for (int j = 0; j < 6; ++j)
            cs[j] = p[j] + p[WS_W + j] + p[2 * WS_W + j];
        f4 out;
        const float* pe = &es[(r + 2) * ES_W + (c0 + 2)];
        out.x = pe[0] * (cs[0] + cs[1] + cs[2]);
        out.y = pe[1] * (cs[1] + cs[2] + cs[3]);
        out.z = pe[2] * (cs[2] + cs[3] + cs[4]);
        out.w = pe[3] * (cs[3] + cs[4] + cs[5]);
        int gr = row0 + r, gc = col0 + c0;
        if (gr < H && gc + 3 < W)
            *(f4*)&y[(size_t)gr * W + gc] = out;   // 16B-aligned: W,col multiples of 4
    }
}

extern "C" void kernel_launch(void* const* d_in, const int* in_sizes, int n_in,
                              void* d_out, int out_size, void* d_ws, size_t ws_size,
                              hipStream_t stream) {
    (void)n_in; (void)out_size; (void)d_ws; (void)ws_size;
    const float* x = (const float*)d_in[0];
    const float* k = (const float*)d_in[1];
    float* y = (float*)d_out;

    const int H = (int)lround(sqrt((double)in_sizes[0]));  // 4096
    const int W = H;

    dim3 grid((W + TILE - 1) / TILE, (H + TILE - 1) / TILE);
    energy_dist_fused_kernel<<<grid, 256, 0, stream>>>(x, k, y, H, W);
}